// KernelSelfAttn_25761213841965
// MI455X (gfx1250) — compile-verified
//
#include <hip/hip_runtime.h>
#include <hip/hip_bf16.h>

// N=30000, E=480000, DIM=DQK=128, H=8, FD=DVH=16
// Bandwidth-bound layer (edge scatter ~2GB traffic @ 23.3TB/s). GEMMs kept in
// full fp32 via V_WMMA_F32_16X16X4_F32 (no reason to quantize on a BW-bound op).

typedef __attribute__((ext_vector_type(2))) float v2f;
typedef __attribute__((ext_vector_type(8))) float v8f;

#define KDIM 128

// ---------------------------------------------------------------- zero scratch
__global__ __launch_bounds__(256) void eqt_zero_kernel(float* __restrict__ p, int n) {
  int i = blockIdx.x * 256 + threadIdx.x;
  if (i < n) p[i] = 0.f;
}

// ---------------------------------------------------------------- WMMA f32 GEMM
// C[M x Nout] = A[M x 128] @ B[128 x Nout] (+ bias). One 16x16 tile per wave32.
__global__ __launch_bounds__(256) void eqt_gemm_wmma_kernel(
    const float* __restrict__ A, const float* __restrict__ B,
    const float* __restrict__ bias, float* __restrict__ C,
    int M, int Nout)
{
  const int lane   = threadIdx.x & 31;
  const int wave   = threadIdx.x >> 5;
  const int tilesN = Nout >> 4;
  const int nTiles = (M >> 4) * tilesN;
  const int tile   = blockIdx.x * 8 + wave;
  if (tile >= nTiles) return;               // wave-uniform: EXEC stays all-ones
  const int tm    = tile / tilesN;
  const int tn    = tile - tm * tilesN;
  const int l15   = lane & 15;
  const int khalf = (lane >> 4) << 1;       // 0 for lanes 0-15, 2 for lanes 16-31

  const float* Ap = A + (size_t)(tm * 16 + l15) * KDIM + khalf;
  const float* Bp = B + (size_t)khalf * Nout + tn * 16 + l15;

  v8f acc = {0.f, 0.f, 0.f, 0.f, 0.f, 0.f, 0.f, 0.f};
#pragma unroll 8
  for (int k = 0; k < KDIM; k += 4) {
    // A 16x4 frag: lane m<16 -> (M=m, K=k..k+1); lane 16+m -> (M=m, K=k+2..k+3)
    v2f a = *(const v2f*)(Ap + k);
    // B 4x16 frag mirrored across the K halves
    v2f b;
    b.x = Bp[(size_t)k * Nout];
    b.y = Bp[(size_t)(k + 1) * Nout];
    acc = __builtin_amdgcn_wmma_f32_16x16x4_f32(false, a, false, b,
                                                (short)0, acc, false, false);
  }
  // C/D 16x16 layout: vgpr i, lanes 0-15 -> M=i; lanes 16-31 -> M=8+i
  const int rbase = tm * 16 + ((lane >> 4) << 3);
  const int col   = tn * 16 + l15;
  const float bv  = bias ? bias[col] : 0.f;
#pragma unroll
  for (int i = 0; i < 8; ++i)
    C[(size_t)(rbase + i) * Nout + col] = acc[i] + bv;
}

// ------------------------------------------------- phi(k) outer-product kv / z
// kvz[0:2048] = kv[h][f][d], kvz[2048:2176] = z[h][f]
__global__ __launch_bounds__(256) void eqt_phikv_kernel(
    const float* __restrict__ h1, const float* __restrict__ pv,
    float* __restrict__ kvz, int Nn)
{
  __shared__ float skf[128];
  __shared__ float svh[128];
  const int t     = threadIdx.x;
  const int h     = t >> 5;
  const int f     = (t >> 1) & 15;
  const int dbase = (t & 1) << 3;
  float acc[8] = {0.f, 0.f, 0.f, 0.f, 0.f, 0.f, 0.f, 0.f};
  float zacc = 0.f;

  for (int n = blockIdx.x; n < Nn; n += gridDim.x) {
    if (t < 128) {                           // phi(k) = elu(k)+1
      float kz = h1[(size_t)n * 384 + 128 + t];
      skf[t] = (kz > 0.f) ? (kz + 1.f) : __expf(kz);
    } else {
      svh[t - 128] = pv[(size_t)n * 384 + (t - 128)];
    }
    __syncthreads();
    const float kfv = skf[(h << 4) + f];
#pragma unroll
    for (int j = 0; j < 8; ++j)
      acc[j] = fmaf(kfv, svh[(h << 4) + dbase + j], acc[j]);
    if (!(t & 1)) zacc += kfv;
    __syncthreads();
  }
#pragma unroll
  for (int j = 0; j < 8; ++j)
    atomicAdd(&kvz[(h << 8) + (f << 4) + dbase + j], acc[j]);
  if (!(t & 1)) atomicAdd(&kvz[2048 + (h << 4) + f], zacc);
}

// ----------------------------------------------------- att = (qf@kv)/(qf@z+eps)
__global__ __launch_bounds__(128) void eqt_att_kernel(
    const float* __restrict__ h1, const float* __restrict__ kvz,
    float* __restrict__ att, int Nn)
{
  __shared__ float sq[128];
  const int n = blockIdx.x;
  const int d = threadIdx.x;
  float qz = h1[(size_t)n * 384 + d];
  sq[d] = (qz > 0.f) ? (qz + 1.f) : __expf(qz);
  __syncthreads();
  const int h = d >> 4, dh = d & 15;
  float num = 0.f, den = 0.f;
#pragma unroll
  for (int f = 0; f < 16; ++f) {
    float qf = sq[(h << 4) + f];
    num = fmaf(qf, kvz[(h << 8) + (f << 4) + dh], num);
    den = fmaf(qf, kvz[2048 + (h << 4) + f], den);
  }
  att[(size_t)n * 128 + d] = num / (den + 1e-6f);
}

// --------------------------- vp = vec@W_vecproj; vec_dot = sum_c v1p*v2p; + v3p
__global__ __launch_bounds__(128) void eqt_vecproj_kernel(
    const float* __restrict__ vec, const float* __restrict__ Wvp,
    float* __restrict__ vec_dot, float* __restrict__ v3p, int Nn)
{
  __shared__ float sv[384];
  const int n = blockIdx.x;
  const int d = threadIdx.x;
#pragma unroll
  for (int c = 0; c < 3; ++c)
    sv[c * 128 + d] = vec[(size_t)n * 384 + c * 128 + d];
  __syncthreads();
  float vdot = 0.f;
#pragma unroll
  for (int c = 0; c < 3; ++c) {
    float a1 = 0.f, a2 = 0.f, a3 = 0.f;
#pragma unroll 16
    for (int i = 0; i < 128; ++i) {
      float vv = sv[c * 128 + i];
      a1 = fmaf(vv, Wvp[(size_t)i * 384 + d], a1);
      a2 = fmaf(vv, Wvp[(size_t)i * 384 + 128 + d], a2);
      a3 = fmaf(vv, Wvp[(size_t)i * 384 + 256 + d], a3);
    }
    vdot = fmaf(a1, a2, vdot);
    v3p[(size_t)n * 384 + c * 128 + d] = a3;
  }
  vec_dot[(size_t)n * 128 + d] = vdot;
}

// ----------------------------------------------------- edge message scatter-add
__global__ __launch_bounds__(256) void eqt_edge_kernel(
    const int* __restrict__ eidx, const float* __restrict__ vec,
    const float* __restrict__ pv, const float* __restrict__ edge_vec,
    float* __restrict__ vec_agg, int Ee)
{
  int tid = blockIdx.x * 256 + threadIdx.x;
  if (tid >= Ee * 128) return;
  const int e = tid >> 7;
  const int d = tid & 127;
  const int src = eidx[e];
  const int dst = eidx[Ee + e];
  const float v1 = pv[(size_t)src * 384 + 128 + d];   // vec1
  const float v2 = pv[(size_t)src * 384 + 256 + d];   // vec2
#pragma unroll
  for (int c = 0; c < 3; ++c) {
    float m = fmaf(vec[(size_t)src * 384 + c * 128 + d], v1,
                   edge_vec[(size_t)e * 3 + c] * v2);
    atomicAdd(&vec_agg[(size_t)dst * 384 + c * 128 + d], m);
  }
}

// ------------------------------------- t2 = att + vec_dot*o2 + o3 (into vec_dot)
__global__ __launch_bounds__(256) void eqt_dx_kernel(
    const float* __restrict__ att, const float* __restrict__ o,
    float* __restrict__ vec_dot, int Nn)
{
  int idx = blockIdx.x * 256 + threadIdx.x;
  if (idx >= Nn * 128) return;
  const int n = idx >> 7, d = idx & 127;
  const float o2 = o[(size_t)n * 384 + 128 + d];
  const float o3 = o[(size_t)n * 384 + 256 + d];
  vec_dot[idx] = att[idx] + fmaf(vec_dot[idx], o2, o3);
}

// --------------------------------- out_x = non_att + fco_out * edge_weight[n]
__global__ __launch_bounds__(256) void eqt_outx_kernel(
    const float* __restrict__ h1, const float* __restrict__ fco,
    const float* __restrict__ edge_weight, float* __restrict__ out_x, int Nn)
{
  int idx = blockIdx.x * 256 + threadIdx.x;
  if (idx >= Nn * 128) return;
  const int n = idx >> 7, d = idx & 127;
  out_x[idx] = h1[(size_t)n * 384 + 256 + d] + fco[idx] * edge_weight[n];
}

// ------------- delta_vec = v3p*o1 + vec_agg; layernorm over d; write into v3p
__global__ __launch_bounds__(128) void eqt_ln_kernel(
    const float* __restrict__ o, const float* __restrict__ vec_agg,
    const float* __restrict__ ln_scale, const float* __restrict__ ln_bias,
    float* __restrict__ v3p, int Nn)
{
  __shared__ float s1[128];
  __shared__ float s2[128];
  const int r = blockIdx.x;          // row = n*3 + c, contiguous (N,3,128)
  const int n = r / 3;
  const int d = threadIdx.x;
  const float o1 = o[(size_t)n * 384 + d];
  const float dv = fmaf(v3p[(size_t)r * 128 + d], o1, vec_agg[(size_t)r * 128 + d]);
  s1[d] = dv;
  s2[d] = dv * dv;
  __syncthreads();
  for (int s = 64; s > 0; s >>= 1) {
    if (d < s) { s1[d] += s1[d + s]; s2[d] += s2[d + s]; }
    __syncthreads();
  }
  const float mu  = s1[0] * (1.f / 128.f);
  const float var = s2[0] * (1.f / 128.f) - mu * mu;
  const float inv = rsqrtf(var + 1e-5f);
  v3p[(size_t)r * 128 + d] = fmaf((dv - mu) * inv, ln_scale[d], ln_bias[d]);
}

// =============================================================================
extern "C" void kernel_launch(void* const* d_in, const int* in_sizes, int n_in,
                              void* d_out, int out_size, void* d_ws, size_t ws_size,
                              hipStream_t stream)
{
  const float* x          = (const float*)d_in[0];
  const float* vec        = (const float*)d_in[1];
  const float* edge_vec   = (const float*)d_in[2];
  const float* edge_weight= (const float*)d_in[3];
  const float* W_fc1      = (const float*)d_in[4];
  const float* W_fcv      = (const float*)d_in[5];
  const float* b_fcv      = (const float*)d_in[6];
  const float* W_fco      = (const float*)d_in[7];
  const float* b_fco      = (const float*)d_in[8];
  const float* W_out      = (const float*)d_in[9];
  const float* W_vecproj  = (const float*)d_in[10];
  const float* W_equiv    = (const float*)d_in[11];
  const float* ln_scale   = (const float*)d_in[12];
  const float* ln_bias    = (const float*)d_in[13];
  const int*   edge_index = (const int*)d_in[14];

  const int Nn = in_sizes[0] / 128;     // 30000
  const int Ee = in_sizes[14] / 2;      // 480000

  // workspace layout (floats)
  float* ws = (float*)d_ws;
  size_t off = 0;
  float* h1   = ws + off; off += (size_t)Nn * 384;   // q|k|non_att
  float* pv   = ws + off; off += (size_t)Nn * 384;   // v|vec1|vec2
  float* att  = ws + off; off += (size_t)Nn * 128;   // att, later fco result
  float* kvz  = ws + off; off += 4096;               // kv(2048) + z(128)
  float* vdot = ws + off; off += (size_t)Nn * 128;   // vec_dot, later t2
  float* v3p  = ws + off; off += (size_t)Nn * 384;   // v3p, later dvn
  float* vagg = ws + off; off += (size_t)Nn * 384;   // segment_sum accumulator
  float* ob   = ws + off; off += (size_t)Nn * 384;   // o = att@W_out

  float* out_x   = (float*)d_out;
  float* out_vec = out_x + (size_t)Nn * 128;

  auto gemm = [&](const float* A, const float* B, const float* bias, float* C,
                  int M, int Nout) {
    int nTiles = (M >> 4) * (Nout >> 4);
    eqt_gemm_wmma_kernel<<<(nTiles + 7) / 8, 256, 0, stream>>>(A, B, bias, C, M, Nout);
  };

  // 0) zero the atomic accumulators (every call: graph-replay deterministic)
  eqt_zero_kernel<<<(4096 + 255) / 256, 256, 0, stream>>>(kvz, 4096);
  {
    int nz = Nn * 384;
    eqt_zero_kernel<<<(nz + 255) / 256, 256, 0, stream>>>(vagg, nz);
  }

  // 1) h1 = x @ W_fc1 ; pv = x @ W_fcv + b_fcv        (WMMA f32)
  gemm(x, W_fc1, nullptr, h1, Nn, 384);
  gemm(x, W_fcv, b_fcv,   pv, Nn, 384);

  // 2) kv/z accumulation (linear attention state)
  eqt_phikv_kernel<<<256, 256, 0, stream>>>(h1, pv, kvz, Nn);

  // 3) att = (qf@kv)/(qf@z + eps)
  eqt_att_kernel<<<Nn, 128, 0, stream>>>(h1, kvz, att, Nn);

  // 4) vec projection -> vec_dot, v3p
  eqt_vecproj_kernel<<<Nn, 128, 0, stream>>>(vec, W_vecproj, vdot, v3p, Nn);

  // 5) edge message scatter (dominant cost: ~2GB traffic + f32 atomics)
  {
    int nthreads = Ee * 128;
    eqt_edge_kernel<<<(nthreads + 255) / 256, 256, 0, stream>>>(
        edge_index, vec, pv, edge_vec, vagg, Ee);
  }

  // 6) o = att @ W_out                                (WMMA f32)
  gemm(att, W_out, nullptr, ob, Nn, 384);

  // 7) t2 = att + delta_x  (into vdot)
  eqt_dx_kernel<<<(Nn * 128 + 255) / 256, 256, 0, stream>>>(att, ob, vdot, Nn);

  // 8) fco = t2 @ W_fco + b_fco  (into att buffer)    (WMMA f32)
  gemm(vdot, W_fco, b_fco, att, Nn, 128);

  // 9) out_x = non_att + fco * edge_weight
  eqt_outx_kernel<<<(Nn * 128 + 255) / 256, 256, 0, stream>>>(h1, att, edge_weight, out_x, Nn);

  // 10) delta_vec + layernorm (into v3p)
  eqt_ln_kernel<<<Nn * 3, 128, 0, stream>>>(ob, vagg, ln_scale, ln_bias, v3p, Nn);

  // 11) out_vec = dvn @ W_equiv                       (WMMA f32, M = 3N)
  gemm(v3p, W_equiv, nullptr, out_vec, Nn * 3, 128);
}